// NLPSN_24111946400218
// MI455X (gfx1250) — compile-verified
//
#include <hip/hip_runtime.h>
#include <math.h>

typedef float v2f __attribute__((ext_vector_type(2)));
typedef float v8f __attribute__((ext_vector_type(8)));

#define H_  240
#define W_  1216
#define B_  4
#define CG_ 64

// ---------------------------------------------------------------------------
// helpers: bilinear with zero padding (per-corner validity, mmcv-style)
// ---------------------------------------------------------------------------
__device__ __forceinline__ float fetch_zero(const float* __restrict__ img, int yy, int xx) {
    if (yy < 0 || yy >= H_ || xx < 0 || xx >= W_) return 0.0f;
    return img[(size_t)yy * W_ + xx];
}

__device__ __forceinline__ float bilin_zero(const float* __restrict__ img, float sy, float sx) {
    float y0f = floorf(sy), x0f = floorf(sx);
    float wy = sy - y0f, wx = sx - x0f;
    int y0 = (int)y0f, x0 = (int)x0f;
    float v00 = fetch_zero(img, y0,     x0);
    float v01 = fetch_zero(img, y0,     x0 + 1);
    float v10 = fetch_zero(img, y0 + 1, x0);
    float v11 = fetch_zero(img, y0 + 1, x0 + 1);
    return v00 * (1.0f - wy) * (1.0f - wx) + v01 * (1.0f - wy) * wx +
           v10 * wy * (1.0f - wx)          + v11 * wy * wx;
}

// masked fetch: feat with fixed pixels substituted (mask = feat_fix > 0)
__device__ __forceinline__ float fetch_masked(const float* __restrict__ feat,
                                              const float* __restrict__ ffix,
                                              int yy, int xx) {
    if (yy < 0 || yy >= H_ || xx < 0 || xx >= W_) return 0.0f;
    size_t i = (size_t)yy * W_ + xx;
    float fx = ffix[i];
    return fx > 0.0f ? fx : feat[i];
}

// async global->LDS copy of one dword (GVS mode: sgpr base + 32-bit voffset)
__device__ __forceinline__ void async_g2l_b32(unsigned lds_byte_addr,
                                              unsigned gbyte_off,
                                              const float* base) {
    asm volatile("global_load_async_to_lds_b32 %0, %1, %2"
                 :: "v"(lds_byte_addr), "v"(gbyte_off), "s"(base) : "memory");
}

__device__ __forceinline__ void wait_asynccnt0() {
    asm volatile("s_wait_asynccnt 0x0" ::: "memory");
}

// ---------------------------------------------------------------------------
// Kernel 1: fused 3x3 conv (WMMA f32 implicit GEMM) + offset/affinity epilogue
//   block = 128 threads (4 waves); wave w computes 16 pixels x 24 channels
//   grid  = B * H * (W/64)
// GEMM: M=16 pixels, N=24 (two 16-wide tiles), K=576 ordered k = (ci/16)*144 + t*16 + ci%16
// LDS tile layout: glds[rr][ci][col], col stride 128 (cols 0..65 valid)
// ---------------------------------------------------------------------------
__launch_bounds__(128)
__global__ void nlspn_conv_aff(const float* __restrict__ guidance,   // [B,64,H,W]
                               const float* __restrict__ conv_w,     // [24,64,3,3]
                               const float* __restrict__ conv_b,     // [24]
                               const float* __restrict__ confidence, // [B,1,H,W]
                               const float* __restrict__ aff_scale,  // [1]
                               float* __restrict__ out_offset,       // [B,18,H,W]
                               float* __restrict__ out_aff9)         // [B,9,H,W]
{
    __shared__ float glds[3 * 16 * 128];    // 3 rows x 16 cin x 128-padded cols
    __shared__ float oalds[4][16][25];      // de-stripe scratch: wave x pixel x ch

    const int tid  = threadIdx.x;
    const int wave = tid >> 5;
    const int lane = tid & 31;
    const int half = lane >> 4;             // 0: lanes 0-15, 1: lanes 16-31
    const int l16  = lane & 15;

    int blk = blockIdx.x;
    const int tb = blk % 19; blk /= 19;     // 19 x 64-pixel tiles per row
    const int y  = blk % H_; blk /= H_;
    const int b  = blk;
    const int xblk = tb * 64;

    v8f acc0 = {};                          // channels 0..15
    v8f acc1 = {};                          // channels 16..23 (upper 8 padded)

    const int  co1 = 16 + l16;
    const bool v1  = (co1 < 24);
    const float* w0 = conv_w + (size_t)l16 * 576;
    const float* w1 = conv_w + (size_t)(v1 ? co1 : 0) * 576;

    for (int chunk = 0; chunk < 4; ++chunk) {
        // ---- async fill of guidance chunk (zero padded at image borders) ----
        // main region: 48 (rr,ci)-rows x 64 cols; all indexing is shifts/ands
        for (int e = tid; e < 48 * 64; e += 128) {
            const int row = e >> 6;               // 0..47
            const int rr  = row >> 4;             // 0..2
            const int ci  = row & 15;             // 0..15
            const int col = e & 63;               // 0..63
            const int gy  = y + rr - 1;
            const int gx  = xblk - 1 + col;
            const int li  = (rr << 11) + (ci << 7) + col;
            if (gy >= 0 && gy < H_ && gx >= 0 && gx < W_) {
                unsigned goff = (unsigned)(((((size_t)(b * CG_ + chunk * 16 + ci)) * H_ + gy) * W_ + gx) * 4u);
                async_g2l_b32((unsigned)(size_t)(const void*)&glds[li], goff, guidance);
            } else {
                glds[li] = 0.0f;
            }
        }
        // halo region: cols 64,65 for each of the 48 rows (96 elements)
        if (tid < 96) {
            const int row = tid >> 1;
            const int rr  = row >> 4;
            const int ci  = row & 15;
            const int col = 64 + (tid & 1);
            const int gy  = y + rr - 1;
            const int gx  = xblk - 1 + col;
            const int li  = (rr << 11) + (ci << 7) + col;
            if (gy >= 0 && gy < H_ && gx >= 0 && gx < W_) {
                unsigned goff = (unsigned)(((((size_t)(b * CG_ + chunk * 16 + ci)) * H_ + gy) * W_ + gx) * 4u);
                async_g2l_b32((unsigned)(size_t)(const void*)&glds[li], goff, guidance);
            } else {
                glds[li] = 0.0f;
            }
        }
        wait_asynccnt0();
        __syncthreads();

        // ---- WMMA inner loop: K-steps of 4 (one tap, four cin) ----
        const int colbase = 16 * wave + l16;
        for (int t = 0; t < 9; ++t) {               // tap (kh,kw) — scalar
            const int kh = t / 3, kw = t % 3;
            const int aoff = (kh << 11) + colbase + kw;
            #pragma unroll
            for (int cs = 0; cs < 4; ++cs) {        // 4 K-steps of 4 cin each
                const int cb = cs * 4 + half * 2;   // lanes>=16 carry K+2/K+3
                v2f a;
                a.x = glds[((cb + 0) << 7) + aoff];
                a.y = glds[((cb + 1) << 7) + aoff];
                const int kidx = (chunk * 16 + cb) * 9 + t;  // w[co][ci][t]
                v2f b0, b1;
                b0.x = w0[kidx];
                b0.y = w0[kidx + 9];
                b1.x = v1 ? w1[kidx]     : 0.0f;
                b1.y = v1 ? w1[kidx + 9] : 0.0f;
                acc0 = __builtin_amdgcn_wmma_f32_16x16x4_f32(
                           false, a, false, b0, (short)0, acc0, false, false);
                acc1 = __builtin_amdgcn_wmma_f32_16x16x4_f32(
                           false, a, false, b1, (short)0, acc1, false, false);
            }
        }
        __syncthreads();
    }

    // de-stripe accumulators: VGPR r holds M = r + half*8, N = l16
    #pragma unroll
    for (int r = 0; r < 8; ++r) {
        oalds[wave][half * 8 + r][l16] = acc0[r];
        if (v1) oalds[wave][half * 8 + r][co1] = acc1[r];
    }
    __syncthreads();

    // ---- per-pixel epilogue: one thread per pixel (64 pixels / block) ----
    if (tid < 64) {
        const int wv = tid >> 4;
        const int m  = tid & 15;
        const int x  = xblk + tid;

        float oa[24];
        #pragma unroll
        for (int c = 0; c < 24; ++c) oa[c] = oalds[wv][m][c] + conv_b[c];

        // dy/dx with zero inserted at reference tap (t==4)
        float dy9[9], dx9[9];
        #pragma unroll
        for (int t = 0; t < 9; ++t) {
            if (t == 4)      { dy9[t] = 0.0f;       dx9[t] = 0.0f; }
            else if (t < 4)  { dy9[t] = oa[t];      dx9[t] = oa[8 + t]; }
            else             { dy9[t] = oa[t - 1];  dx9[t] = oa[8 + t - 1]; }
        }

        const float inv_scale = 1.0f / (aff_scale[0] + 1e-8f);
        const float* conf = confidence + (size_t)b * H_ * W_;

        float af[8];
        float s = 0.0f;
        #pragma unroll
        for (int j = 0; j < 8; ++j) {
            const int t = (j < 4) ? j : j + 1;
            float a = tanhf(oa[16 + j]) * inv_scale;
            a *= bilin_zero(conf, (float)y + dy9[t], (float)x + dx9[t]);
            af[j] = a;
            s += fabsf(a);
        }
        s = fmaxf(s + 1e-4f, 1.0f);
        const float invs = 1.0f / s;
        float sum_af = 0.0f;
        #pragma unroll
        for (int j = 0; j < 8; ++j) { af[j] *= invs; sum_af += af[j]; }
        const float aff_ref = 1.0f - sum_af;

        const size_t hw = (size_t)H_ * W_;
        const size_t p  = (size_t)y * W_ + x;
        float* offp = out_offset + (size_t)b * 18 * hw + p;
        float* affp = out_aff9   + (size_t)b * 9  * hw + p;
        #pragma unroll
        for (int t = 0; t < 9; ++t) {
            offp[(size_t)(2 * t)     * hw] = dy9[t];
            offp[(size_t)(2 * t + 1) * hw] = dx9[t];
            float a9 = (t == 4) ? aff_ref : (t < 4 ? af[t] : af[t - 1]);
            affp[(size_t)t * hw] = a9;
        }
    }
}

// ---------------------------------------------------------------------------
// Kernel 2: one propagation step (launched PROP_TIME=6 times, ping-pong)
//   feat_masked(q) = feat_fix(q) > 0 ? feat_fix(q) : feat(q)   (fused in gather)
//   out(p) = sum_t aff9[t](p) * bilinear(feat_masked, y+by[t]+dy[t], x+bx[t]+dx[t])
// ---------------------------------------------------------------------------
__launch_bounds__(256)
__global__ void nlspn_prop(const float* __restrict__ feat_in,   // [B,1,H,W]
                           const float* __restrict__ feat_fix,  // [B,1,H,W]
                           const float* __restrict__ offset,    // [B,18,H,W]
                           const float* __restrict__ aff9,      // [B,9,H,W]
                           float* __restrict__ feat_out)        // [B,1,H,W]
{
    const size_t hw  = (size_t)H_ * W_;
    const size_t gid = (size_t)blockIdx.x * 256 + threadIdx.x;
    if (gid >= (size_t)B_ * hw) return;

    const int    b = (int)(gid / hw);
    const size_t p = gid % hw;
    const int    y = (int)(p / W_);
    const int    x = (int)(p % W_);

    const float* fin = feat_in  + (size_t)b * hw;
    const float* ffx = feat_fix + (size_t)b * hw;
    const float* off = offset   + (size_t)b * 18 * hw + p;
    const float* af  = aff9     + (size_t)b * 9  * hw + p;

    float acc = 0.0f;
    #pragma unroll
    for (int t = 0; t < 9; ++t) {
        const float dy = off[(size_t)(2 * t)     * hw];
        const float dx = off[(size_t)(2 * t + 1) * hw];
        const float a  = af[(size_t)t * hw];
        const float sy = (float)(y + (t / 3) - 1) + dy;
        const float sx = (float)(x + (t % 3) - 1) + dx;

        float y0f = floorf(sy), x0f = floorf(sx);
        float wy = sy - y0f, wx = sx - x0f;
        int   y0 = (int)y0f, x0 = (int)x0f;
        float v00 = fetch_masked(fin, ffx, y0,     x0);
        float v01 = fetch_masked(fin, ffx, y0,     x0 + 1);
        float v10 = fetch_masked(fin, ffx, y0 + 1, x0);
        float v11 = fetch_masked(fin, ffx, y0 + 1, x0 + 1);
        acc += a * (v00 * (1.0f - wy) * (1.0f - wx) + v01 * (1.0f - wy) * wx +
                    v10 * wy * (1.0f - wx)          + v11 * wy * wx);
    }
    feat_out[gid] = acc;
}

// ---------------------------------------------------------------------------
// launch
// ---------------------------------------------------------------------------
extern "C" void kernel_launch(void* const* d_in, const int* in_sizes, int n_in,
                              void* d_out, int out_size, void* d_ws, size_t ws_size,
                              hipStream_t stream) {
    const float* feat_init  = (const float*)d_in[0];
    const float* guidance   = (const float*)d_in[1];
    const float* confidence = (const float*)d_in[2];
    const float* feat_fix   = (const float*)d_in[3];
    const float* conv_w     = (const float*)d_in[4];
    const float* conv_b     = (const float*)d_in[5];
    const float* aff_scale  = (const float*)d_in[6];

    const size_t hw = (size_t)H_ * W_;
    const size_t n1 = (size_t)B_ * hw;

    float* out       = (float*)d_out;
    float* out_feat  = out;                 // [B,1,H,W]
    float* out_off   = out + n1;            // [B,18,H,W]
    float* out_aff   = out + 19 * n1;       // [B,9,H,W]

    float* bufA = (float*)d_ws;             // ping-pong feature buffers
    float* bufB = bufA + n1;

    // 1) fused WMMA conv + offset/affinity epilogue
    nlspn_conv_aff<<<B_ * H_ * 19, 128, 0, stream>>>(
        guidance, conv_w, conv_b, confidence, aff_scale, out_off, out_aff);

    // 2) PROP_TIME = 6 propagation steps (device-wide sync between iterations)
    const int nblk = (int)((n1 + 255) / 256);
    nlspn_prop<<<nblk, 256, 0, stream>>>(feat_init, feat_fix, out_off, out_aff, bufA);
    nlspn_prop<<<nblk, 256, 0, stream>>>(bufA,      feat_fix, out_off, out_aff, bufB);
    nlspn_prop<<<nblk, 256, 0, stream>>>(bufB,      feat_fix, out_off, out_aff, bufA);
    nlspn_prop<<<nblk, 256, 0, stream>>>(bufA,      feat_fix, out_off, out_aff, bufB);
    nlspn_prop<<<nblk, 256, 0, stream>>>(bufB,      feat_fix, out_off, out_aff, bufA);
    nlspn_prop<<<nblk, 256, 0, stream>>>(bufA,      feat_fix, out_off, out_aff, out_feat);
}